// BoxNetwork_40802189312698
// MI455X (gfx1250) — compile-verified
//
#include <hip/hip_runtime.h>
#include <hip/hip_bf16.h>
#include <math.h>

#define CENTER_SIZE 64
#define LOOP_LEN 50

// Raw IEEE minimumNumber without the fminf sNaN-canonicalization dance.
__device__ __forceinline__ float min_raw(float a, float b) {
    float d;
    asm("v_min_num_f32 %0, %1, %2" : "=v"(d) : "v"(a), "v"(b));
    return d;
}

template <int PATTERN>
__device__ __forceinline__ float swz_min(float v) {
    // group-of-32 xor swizzle: offset[15]=0, and_mask=0x1F, xor_mask in [14:10]
    float o = __int_as_float(__builtin_amdgcn_ds_swizzle(__float_as_int(v), PATTERN));
    return min_raw(v, o);
}

// One wave32 does the whole job: latency-bound gather + min-reduce + scalar epilogue.
__global__ __launch_bounds__(32) void boxnet_loss_kernel(
    const int* __restrict__ index_vec,          // low word of int64 index (value < 2^21)
    const int* __restrict__ neighbor_index_vec, // low word of int64 index
    const float* __restrict__ len_sum_p,        // scalar
    const float* __restrict__ emb,              // [1e6, 64] fp32
    float* __restrict__ out)                    // [1] fp32
{
    const int t = threadIdx.x;  // 0..31, single wave32

    // Wave-uniform scalar loads (SMEM / KMcnt path) — issue all up front so
    // their latency overlaps the vector gathers.
    const unsigned ci = (unsigned)index_vec[0];
    const unsigned ni = (unsigned)neighbor_index_vec[0];
    const float    ls = len_sum_p[0];

    const float* __restrict__ crow = emb + (size_t)ci * CENTER_SIZE;
    const float* __restrict__ nrow = emb + (size_t)ni * CENTER_SIZE;

    // Cover elements 0..49 as {t, t+18}: every lane does BOTH loads
    // unconditionally -> 4 global_load_b32 in one clause, one loadcnt wait.
    // Overlapping coverage (18..31 seen twice) is harmless under min().
    float a0 = crow[t];
    float b0 = nrow[t];
    float a1 = crow[t + 18];
    float b1 = nrow[t + 18];

    float v = min_raw(fabsf(a0 - b0), fabsf(a1 - b1));

    // Wave32 min-reduction: 5 ds_swizzle xor-swaps (all lanes active).
    v = swz_min<0x401F>(v);  // xor 16
    v = swz_min<0x201F>(v);  // xor 8
    v = swz_min<0x101F>(v);  // xor 4
    v = swz_min<0x081F>(v);  // xor 2
    v = swz_min<0x041F>(v);  // xor 1

    // Compute the epilogue in ALL lanes so `ls`'s scalar load is issued in the
    // preamble (overlapped with the gathers), not inside the lane-0 branch.
    const float l1   = fabsf(v - ls);
    const float loss = (v < ls) ? 100.0f * l1 : l1;

    if (t == 0) {
        out[0] = loss;
    }
}

extern "C" void kernel_launch(void* const* d_in, const int* in_sizes, int n_in,
                              void* d_out, int out_size, void* d_ws, size_t ws_size,
                              hipStream_t stream) {
    (void)in_sizes; (void)n_in; (void)out_size; (void)d_ws; (void)ws_size;
    const int*   index_vec          = (const int*)d_in[0];
    const int*   neighbor_index_vec = (const int*)d_in[1];
    const float* len_sum            = (const float*)d_in[2];
    const float* emb                = (const float*)d_in[3];
    float*       out                = (float*)d_out;

    boxnet_loss_kernel<<<1, 32, 0, stream>>>(index_vec, neighbor_index_vec,
                                             len_sum, emb, out);
}